// Cache_23888608100419
// MI455X (gfx1250) — compile-verified
//
#include <hip/hip_runtime.h>

typedef float v2f __attribute__((ext_vector_type(2)));
typedef float v8f __attribute__((ext_vector_type(8)));

#define QL      64
#define BSZ     8
#define CACHE_L 64
#define CACHE_N 512
#define NHID    1024
#define TOPK    4

// -----------------------------------------------------------------------------
// Kernel 1: per (cache slot n, batch b) compute the 64x64 score tile
//   S[m][j] = sum_h q[m,b,h] * keys[n,b,j,h]   (K = 1024)
// with V_WMMA_F32_16X16X4_F32, then reduce max over j and write
//   logits[m][b][n] = max_j S[m][j]
// Block: 256 threads = 8 waves. 4 m-tiles x 4 j-tiles of 16x16;
// wave w -> m-tile (w>>1), j-tiles {2*(w&1), 2*(w&1)+1}.
// -----------------------------------------------------------------------------
__global__ __launch_bounds__(256)
void scores_kernel(const float* __restrict__ q,
                   const float* __restrict__ keys,
                   float* __restrict__ logits) {
    const int n    = blockIdx.x;
    const int b    = blockIdx.y;
    const int tid  = threadIdx.x;
    const int wave = tid >> 5;
    const int lane = tid & 31;

    const int mi        = wave >> 1;           // 0..3
    const int m_base    = mi << 4;             // 0,16,32,48
    const int col_base0 = (wave & 1) << 5;     // 0 or 32
    const int col_base1 = col_base0 + 16;

    const int row   = lane & 15;               // matrix row (A) / column (B) for this lane
    const int khalf = lane >> 4;               // 0: K,K+1   1: K+2,K+3

    // A element (M=m, K=k) lives at q[m*(BSZ*NHID) + b*NHID + k]  (query (QL,1,BSZ,NHID))
    const float* qptr = q + (size_t)(m_base + row) * (BSZ * NHID) + (size_t)b * NHID + 2 * khalf;
    // B element (K=k, N=j) lives at keys[n*(BSZ*CACHE_L*NHID) + b*(CACHE_L*NHID) + j*NHID + k]
    const float* kbase = keys + (size_t)n * (BSZ * CACHE_L * NHID)
                              + (size_t)b * (CACHE_L * NHID) + 2 * khalf;
    const float* kptr0 = kbase + (size_t)(col_base0 + row) * NHID;
    const float* kptr1 = kbase + (size_t)(col_base1 + row) * NHID;

    v8f c0 = {};
    v8f c1 = {};

#pragma unroll 4
    for (int k0 = 0; k0 < NHID; k0 += 4) {
        v2f a  = *(const v2f*)(qptr  + k0);
        v2f b0 = *(const v2f*)(kptr0 + k0);
        v2f b1 = *(const v2f*)(kptr1 + k0);
        // 8 args: (neg_a, A, neg_b, B, c_mod, C, reuse_a, reuse_b)
        c0 = __builtin_amdgcn_wmma_f32_16x16x4_f32(false, a, false, b0, (short)0, c0, false, false);
        c1 = __builtin_amdgcn_wmma_f32_16x16x4_f32(false, a, false, b1, (short)0, c1, false, false);
    }

    // C/D layout: VGPR r -> (M=m_base+r, N=lane) on lanes 0-15,
    //                       (M=m_base+8+r, N=lane-16) on lanes 16-31.
    // Max over this wave's 32 columns: per-lane max of the two tiles, then
    // shfl_xor max within each 16-lane group.
    float rowmax[8];
#pragma unroll
    for (int r = 0; r < 8; ++r) {
        float v = fmaxf(c0[r], c1[r]);
        v = fmaxf(v, __shfl_xor(v, 1, 32));
        v = fmaxf(v, __shfl_xor(v, 2, 32));
        v = fmaxf(v, __shfl_xor(v, 4, 32));
        v = fmaxf(v, __shfl_xor(v, 8, 32));
        rowmax[r] = v;
    }

    __shared__ float smax[2][QL];   // [j-half parity][row]
    const int p = wave & 1;
    if (lane == 0) {
#pragma unroll
        for (int r = 0; r < 8; ++r) smax[p][m_base + r] = rowmax[r];
    } else if (lane == 16) {
#pragma unroll
        for (int r = 0; r < 8; ++r) smax[p][m_base + 8 + r] = rowmax[r];
    }
    __syncthreads();

    if (tid < QL) {
        float mx = fmaxf(smax[0][tid], smax[1][tid]);
        logits[(size_t)tid * (BSZ * CACHE_N) + (size_t)b * CACHE_N + n] = mx;
    }
}

// -----------------------------------------------------------------------------
// Kernel 2: per (ql row m, batch b): softmax over the 512 cache slots
// (scale 1/sqrt(NHID) = 1/32), write attention, then 4 argmax rounds for topk.
// Block: 512 threads, one per cache slot.
// -----------------------------------------------------------------------------
__global__ __launch_bounds__(512)
void softmax_topk_kernel(const float* __restrict__ logits,
                         float* __restrict__ att,
                         float* __restrict__ idx_out) {
    const int m   = blockIdx.x;
    const int b   = blockIdx.y;
    const int tid = threadIdx.x;   // == n

    __shared__ float arr[CACHE_N];
    __shared__ float red_v[CACHE_N];
    __shared__ int   red_i[CACHE_N];

    float x = logits[(size_t)m * (BSZ * CACHE_N) + (size_t)b * CACHE_N + tid] * 0.03125f;

    // ---- max reduce ----
    red_v[tid] = x;
    __syncthreads();
    for (int s = CACHE_N / 2; s > 0; s >>= 1) {
        if (tid < s) red_v[tid] = fmaxf(red_v[tid], red_v[tid + s]);
        __syncthreads();
    }
    float mx = red_v[0];
    __syncthreads();

    // ---- sum reduce ----
    float e = expf(x - mx);
    red_v[tid] = e;
    __syncthreads();
    for (int s = CACHE_N / 2; s > 0; s >>= 1) {
        if (tid < s) red_v[tid] += red_v[tid + s];
        __syncthreads();
    }
    float inv = 1.0f / red_v[0];
    __syncthreads();

    float pr = e * inv;
    att[((size_t)m * BSZ + b) * CACHE_N + tid] = pr;
    arr[tid] = pr;
    __syncthreads();

    // ---- top-4 via repeated argmax (ties -> smaller index, as lax.top_k) ----
    for (int kk = 0; kk < TOPK; ++kk) {
        red_v[tid] = arr[tid];
        red_i[tid] = tid;
        __syncthreads();
        for (int s = CACHE_N / 2; s > 0; s >>= 1) {
            if (tid < s) {
                float ov = red_v[tid + s];
                int   oi = red_i[tid + s];
                if (ov > red_v[tid] || (ov == red_v[tid] && oi < red_i[tid])) {
                    red_v[tid] = ov;
                    red_i[tid] = oi;
                }
            }
            __syncthreads();
        }
        if (tid == 0) {
            int best = red_i[0];
            idx_out[(size_t)kk * (QL * BSZ) + (size_t)m * BSZ + b] = (float)best;
            arr[best] = -1.0f;   // probabilities are >= 0, so this masks it out
        }
        __syncthreads();
    }
}

// -----------------------------------------------------------------------------
extern "C" void kernel_launch(void* const* d_in, const int* in_sizes, int n_in,
                              void* d_out, int out_size, void* d_ws, size_t ws_size,
                              hipStream_t stream) {
    (void)in_sizes; (void)n_in; (void)out_size; (void)ws_size;

    const float* q    = (const float*)d_in[0];   // (64, 1, 8, 1024) f32
    const float* keys = (const float*)d_in[1];   // (512, 8, 65536) f32

    float* out  = (float*)d_out;
    float* att  = out;                                   // (64, 8, 512)
    float* idxo = out + (size_t)QL * BSZ * CACHE_N;      // (4, 64, 8) as floats

    float* logits = (float*)d_ws;                        // (64, 8, 512) f32 = 1 MB scratch

    dim3 gridA(CACHE_N, BSZ);
    scores_kernel<<<gridA, 256, 0, stream>>>(q, keys, logits);

    dim3 gridB(QL, BSZ);
    softmax_topk_kernel<<<gridB, 512, 0, stream>>>(logits, att, idxo);
}